// LigerSparsemax_66288525246733
// MI455X (gfx1250) — compile-verified
//
#include <hip/hip_runtime.h>
#include <stdint.h>

// Sparsemax over rows of length 4096, one 256-thread workgroup per row.
// Row staged into LDS with gfx1250 async global->LDS B128 transfers
// (ASYNCcnt path), then pulled once into registers; Michelot's simplex-
// projection fixed point iterates on registers with wave32 shuffles.

#define ROW_D   4096
#define BLOCK   256
#define NWAVES  (BLOCK / 32)
#define PER_T   (ROW_D / BLOCK)   // 16 elements per thread (4 x float4)

__global__ __launch_bounds__(BLOCK) void
sparsemax_row_kernel(const float* __restrict__ x, float* __restrict__ out, int rows)
{
    __shared__ float srow[ROW_D];          // 16 KB row staging
    __shared__ float red_s[NWAVES];        // per-wave partial sums
    __shared__ int   red_c[NWAVES];        // per-wave partial counts
    __shared__ float tau_sh;               // broadcast threshold

    const int row = blockIdx.x;
    if (row >= rows) return;

    const int t    = threadIdx.x;
    const int lane = t & 31;
    const int wv   = t >> 5;

    const float* __restrict__ gx = x   + (size_t)row * ROW_D;
    float*       __restrict__ gy = out + (size_t)row * ROW_D;

    // ---- Stage row into LDS via async global->LDS (ASYNCcnt path) ----
    // 4 chunks: 256 lanes x 16 B = 4 KB per chunk, 16 KB total.
    {
        const unsigned lds_base = (unsigned)(uintptr_t)(&srow[0]); // low 32b of flat LDS addr = LDS offset
#pragma unroll
        for (int c = 0; c < (ROW_D * 4) / (BLOCK * 16); ++c) {
            unsigned    dst = lds_base + (unsigned)((c * BLOCK + t) * 16);
            const void* src = (const char*)gx + (size_t)((c * BLOCK + t) * 16);
            asm volatile("global_load_async_to_lds_b128 %0, %1, off"
                         :: "v"(dst), "v"(src)
                         : "memory");
        }
        asm volatile("s_wait_asynccnt 0x0" ::: "memory");  // drain this wave's transfers
    }
    __syncthreads();

    // ---- Pull this thread's 16 elements into registers (ds_load_b128 x4) ----
    float4 r[PER_T / 4];
    {
        const float4* __restrict__ s4 = (const float4*)srow;
#pragma unroll
        for (int j = 0; j < PER_T / 4; ++j)
            r[j] = s4[t + BLOCK * j];
    }

    // ---- Michelot fixed point: tau <- (sum_{x_i > tau} x_i - 1) / k ----
    // Start at tau = -inf so iteration 0 computes the full-row sum (k = ROW_D).
    float tau = -INFINITY;
    for (int it = 0; it < 64; ++it) {
        float s = 0.0f;
        int   c = 0;
#pragma unroll
        for (int j = 0; j < PER_T / 4; ++j) {
            if (r[j].x > tau) { s += r[j].x; c += 1; }
            if (r[j].y > tau) { s += r[j].y; c += 1; }
            if (r[j].z > tau) { s += r[j].z; c += 1; }
            if (r[j].w > tau) { s += r[j].w; c += 1; }
        }
        // wave32 tree reduction
#pragma unroll
        for (int off = 16; off > 0; off >>= 1) {
            s += __shfl_down(s, off, 32);
            c += __shfl_down(c, off, 32);
        }
        if (lane == 0) { red_s[wv] = s; red_c[wv] = c; }
        __syncthreads();
        if (t == 0) {
            float S = 0.0f;
            int   K = 0;
#pragma unroll
            for (int w = 0; w < NWAVES; ++w) { S += red_s[w]; K += red_c[w]; }
            tau_sh = (S - 1.0f) / (float)K;  // K >= 1 always (max element stays in support)
        }
        __syncthreads();
        float tn = tau_sh;
        if (!(tn > tau)) break;              // uniform branch: support set converged
        tau = tn;
    }

    // ---- Emit max(x - tau, 0) straight from registers (global_store_b128) ----
    {
        float4* __restrict__ o4 = (float4*)gy;
#pragma unroll
        for (int j = 0; j < PER_T / 4; ++j) {
            float4 v = r[j];
            v.x = fmaxf(v.x - tau, 0.0f);
            v.y = fmaxf(v.y - tau, 0.0f);
            v.z = fmaxf(v.z - tau, 0.0f);
            v.w = fmaxf(v.w - tau, 0.0f);
            o4[t + BLOCK * j] = v;
        }
    }
}

extern "C" void kernel_launch(void* const* d_in, const int* in_sizes, int n_in,
                              void* d_out, int out_size, void* d_ws, size_t ws_size,
                              hipStream_t stream)
{
    (void)n_in; (void)out_size; (void)d_ws; (void)ws_size;
    const float* x   = (const float*)d_in[0];
    float*       out = (float*)d_out;
    const int rows   = in_sizes[0] / ROW_D;   // (4 * 4096 * 4096) / 4096 = 16384
    sparsemax_row_kernel<<<dim3(rows), dim3(BLOCK), 0, stream>>>(x, out, rows);
}